// MultiHeadAttentionLayer_43241730736590
// MI455X (gfx1250) — compile-verified
//
#include <hip/hip_runtime.h>

// ---------------------------------------------------------------------------
// Multi-head graph attention (adjacency-masked) for MI455X / gfx1250.
// Kernel 1: fused Q/K/V projection GEMMs, V_WMMA_F32_16X16X4_F32,
//           2 M-tiles per wave sharing B fragments.
// Kernel 2: fused flash-attention. 4 waves/block share one (batch,head);
//           K/V 16x64 tiles staged in LDS via async global->LDS copies
//           (double buffered, ASYNCcnt) and consumed by WMMA fragments.
// All matrix math stays in f32 to match the f32 reference + exact -inf mask.
// ---------------------------------------------------------------------------

typedef __attribute__((ext_vector_type(2))) float v2f;
typedef __attribute__((ext_vector_type(8))) float v8f;

#define IN_DIM   256
#define OUT_DIM  64
#define NUM_HEADS 8
#define BATCH    8
#define NN       1024
#define QKV_N    (OUT_DIM * NUM_HEADS)   // 512
#define M_TOT    (BATCH * NN)            // 8192

__device__ __forceinline__ v8f wmma_f32(v2f a, v2f b, v8f c) {
    // V_WMMA_F32_16X16X4_F32: D = A(16x4) * B(4x16) + C(16x16)
    return __builtin_amdgcn_wmma_f32_16x16x4_f32(
        false, a, false, b, (short)0, c, false, false);
}

// ---- async global -> LDS (CDNA5 path, ASYNCcnt), with safe fallback -------
#if defined(__gfx1250__) && __has_builtin(__builtin_amdgcn_global_load_async_to_lds_b128)
#define USE_ASYNC_LDS 1
#else
#define USE_ASYNC_LDS 0
#endif

#if USE_ASYNC_LDS
typedef int v4i_t __attribute__((vector_size(16)));
typedef __attribute__((address_space(1))) v4i_t* g_v4i_p;
typedef __attribute__((address_space(3))) v4i_t* l_v4i_p;
#endif

__device__ __forceinline__ void copy_f4_to_lds(const float* __restrict__ g, float* l) {
#if USE_ASYNC_LDS
    __builtin_amdgcn_global_load_async_to_lds_b128(
        (g_v4i_p)g, (l_v4i_p)l, 0, 0);
#else
    *(float4*)l = *(const float4*)g;
#endif
}

__device__ __forceinline__ void wait_lds_copies() {
#if USE_ASYNC_LDS
#if __has_builtin(__builtin_amdgcn_s_wait_asynccnt)
    __builtin_amdgcn_s_wait_asynccnt(0);
#else
    __asm__ volatile("s_wait_asynccnt 0" ::: "memory");
#endif
#endif
}

// ---------------------------------------------------------------------------
// Kernel 1: out = h @ W + b for Wq/Wk/Wv selected by blockIdx.z.
// One wave computes a 32x16 output tile (two 16x16 WMMA tiles sharing the
// B fragment). Grid: (QKV_N/16, M_TOT/(32*8), 3), block 256 (8 waves).
// ---------------------------------------------------------------------------
__global__ void __launch_bounds__(256)
qkv_gemm_kernel(const float* __restrict__ h,
                const float* __restrict__ Wq, const float* __restrict__ bq,
                const float* __restrict__ Wk, const float* __restrict__ bk,
                const float* __restrict__ Wv, const float* __restrict__ bv,
                float* __restrict__ q_out, float* __restrict__ k_out,
                float* __restrict__ v_out)
{
    const int lane = threadIdx.x & 31;
    const int wave = threadIdx.x >> 5;
    const int row  = lane & 15;      // M index for A-frag, N index for B/C-frag
    const int hi   = lane >> 4;      // half-wave select

    const float* W;  const float* bias;  float* out;
    if (blockIdx.z == 0)      { W = Wq; bias = bq; out = q_out; }
    else if (blockIdx.z == 1) { W = Wk; bias = bk; out = k_out; }
    else                      { W = Wv; bias = bv; out = v_out; }

    const int n0 = blockIdx.x * 16;
    const int m0 = (blockIdx.y * 8 + wave) * 32;

    const float* A0 = h + (size_t)(m0 + row) * IN_DIM + 2 * hi;   // M tile 0
    const float* A1 = A0 + 16 * IN_DIM;                           // M tile 1
    const float* Bcol = W + n0 + row;

    v8f acc0 = {}, acc1 = {};
    #pragma unroll 4
    for (int k = 0; k < IN_DIM; k += 4) {
        v2f bfr;
        bfr.x = Bcol[(size_t)(k + 2 * hi + 0) * QKV_N];
        bfr.y = Bcol[(size_t)(k + 2 * hi + 1) * QKV_N];
        v2f a0; a0.x = A0[k]; a0.y = A0[k + 1];
        v2f a1; a1.x = A1[k]; a1.y = A1[k + 1];
        acc0 = wmma_f32(a0, bfr, acc0);
        acc1 = wmma_f32(a1, bfr, acc1);
    }

    const float bias_v = bias[n0 + row];
    float* Op = out + (size_t)m0 * QKV_N + n0 + row;
    #pragma unroll
    for (int v = 0; v < 8; ++v) {
        Op[(size_t)(v + 8 * hi) * QKV_N]      = acc0[v] + bias_v;
        Op[(size_t)(16 + v + 8 * hi) * QKV_N] = acc1[v] + bias_v;
    }
}

// ---------------------------------------------------------------------------
// Kernel 2: fused masked attention, flash style, LDS-staged K/V tiles.
// 4 waves/block, all on the same (batch, head), each owning 16 query rows.
// Per column block: async-copy next K/V tile (double buffer), 16 WMMA for
// S = Q K^T from LDS, online softmax in registers, P transposed through a
// private LDS tile, 16 WMMA for O += P V from LDS.
// Grid: (NN/16/K2_WAVES, NUM_HEADS, BATCH), block 32*K2_WAVES.
// ---------------------------------------------------------------------------
#define K2_WAVES 4
#define KV_STRIDE 68   // 64 + 4 pad: bank stride 4 -> conflict-free column reads

__global__ void __launch_bounds__(32 * K2_WAVES)
attention_kernel(const float* __restrict__ Q, const float* __restrict__ K,
                 const float* __restrict__ V, const int* __restrict__ adj,
                 float* __restrict__ out)
{
    __shared__ __align__(16) float ktile[2][16][KV_STRIDE];
    __shared__ __align__(16) float vtile[2][16][KV_STRIDE];
    __shared__ float pbuf[K2_WAVES][16][20];

    const int tid  = threadIdx.x;
    const int lane = tid & 31;
    const int wave = tid >> 5;
    const int row  = lane & 15;
    const int hi   = lane >> 4;

    const int b  = blockIdx.z;
    const int hd = blockIdx.y;
    const int n0 = (blockIdx.x * K2_WAVES + wave) * 16;

    const size_t base = ((size_t)b * NN) * QKV_N + (size_t)hd * OUT_DIM;
    const float* Kbh = K + base;
    const float* Vbh = V + base;
    const float NEG_INF = -__builtin_inff();

    // Cooperative tile loader: 128 threads x 2 float4 = 16x64 floats per tile.
    auto load_tiles = [&](int mb, int buf) {
        #pragma unroll
        for (int i = 0; i < 2; ++i) {
            const int f = tid + i * 128;       // float4 slot 0..255
            const int r = f >> 4;              // 0..15
            const int c = (f & 15) << 2;       // 0..60
            copy_f4_to_lds(Kbh + (size_t)(mb + r) * QKV_N + c, &ktile[buf][r][c]);
            copy_f4_to_lds(Vbh + (size_t)(mb + r) * QKV_N + c, &vtile[buf][r][c]);
        }
    };

    load_tiles(0, 0);   // prefetch first tile pair

    // Q A-fragments for all 16 K-steps over D=64 (kept resident: 32 VGPRs)
    v2f qf[16];
    {
        const float* Qr = Q + base + (size_t)(n0 + row) * QKV_N + 2 * hi;
        #pragma unroll
        for (int t = 0; t < 16; ++t) { qf[t].x = Qr[4 * t]; qf[t].y = Qr[4 * t + 1]; }
    }

    float mrow[8], lrow[8];
    #pragma unroll
    for (int v = 0; v < 8; ++v) { mrow[v] = NEG_INF; lrow[v] = 0.0f; }
    v8f o0 = {}, o1 = {}, o2 = {}, o3 = {};

    const int* adjr = adj + (size_t)(n0 + 8 * hi) * NN;

    for (int j = 0; j < NN / 16; ++j) {
        const int cur = j & 1;
        const int mb  = j * 16;

        wait_lds_copies();     // tiles[cur] landed (per-wave ASYNCcnt)
        __syncthreads();       // ... and visible to all waves
        if (j + 1 < NN / 16)
            load_tiles(mb + 16, cur ^ 1);   // overlap next tile DMA with math

        // ---- S(16x16) = Q(16x64) * K^T(64x16), B-frags from LDS ----
        v8f s = {};
        #pragma unroll
        for (int t = 0; t < 16; ++t) {
            v2f kf;
            kf.x = ktile[cur][row][4 * t + 2 * hi + 0];
            kf.y = ktile[cur][row][4 * t + 2 * hi + 1];
            s = wmma_f32(qf[t], kf, s);
        }

        // ---- mask + online softmax (rows live in 16-lane halves) ----
        float scale[8];
        #pragma unroll
        for (int v = 0; v < 8; ++v) {
            float sv = s[v] * 0.125f;                    // 1/sqrt(64)
            if (adjr[(size_t)v * NN + mb + row] == 0) sv = NEG_INF;

            float mx = sv;
            mx = fmaxf(mx, __shfl_xor(mx, 1, 32));
            mx = fmaxf(mx, __shfl_xor(mx, 2, 32));
            mx = fmaxf(mx, __shfl_xor(mx, 4, 32));
            mx = fmaxf(mx, __shfl_xor(mx, 8, 32));

            const float mnew = fmaxf(mrow[v], mx);
            const float sc   = (mrow[v] == mnew) ? 1.0f : __expf(mrow[v] - mnew);
            const float p    = (sv == NEG_INF) ? 0.0f : __expf(sv - mnew);

            float ps = p;
            ps += __shfl_xor(ps, 1, 32);
            ps += __shfl_xor(ps, 2, 32);
            ps += __shfl_xor(ps, 4, 32);
            ps += __shfl_xor(ps, 8, 32);

            lrow[v]  = lrow[v] * sc + ps;
            mrow[v]  = mnew;
            scale[v] = sc;
            pbuf[wave][v + 8 * hi][row] = p;             // C-layout -> LDS
        }
        #pragma unroll
        for (int v = 0; v < 8; ++v) {
            o0[v] *= scale[v]; o1[v] *= scale[v];
            o2[v] *= scale[v]; o3[v] *= scale[v];
        }
        __syncthreads();   // pbuf writes visible; also orders tile reuse

        // ---- O(16x64) += P(16x16) * V(16x64); P read back in A-layout ----
        #pragma unroll
        for (int t = 0; t < 4; ++t) {
            v2f pf;
            pf.x = pbuf[wave][row][4 * t + 2 * hi + 0];
            pf.y = pbuf[wave][row][4 * t + 2 * hi + 1];
            const int r0 = 4 * t + 2 * hi;
            v2f vf;
            vf.x = vtile[cur][r0][row];       vf.y = vtile[cur][r0 + 1][row];       o0 = wmma_f32(pf, vf, o0);
            vf.x = vtile[cur][r0][row + 16];  vf.y = vtile[cur][r0 + 1][row + 16];  o1 = wmma_f32(pf, vf, o1);
            vf.x = vtile[cur][r0][row + 32];  vf.y = vtile[cur][r0 + 1][row + 32];  o2 = wmma_f32(pf, vf, o2);
            vf.x = vtile[cur][r0][row + 48];  vf.y = vtile[cur][r0 + 1][row + 48];  o3 = wmma_f32(pf, vf, o3);
        }
    }

    // ---- normalize and write out[b][n][hd*64 + d] ----
    #pragma unroll
    for (int v = 0; v < 8; ++v) {
        const float inv = 1.0f / lrow[v];
        float* Or = out + ((size_t)(b * NN + n0 + v + 8 * hi)) * QKV_N
                        + (size_t)hd * OUT_DIM + row;
        Or[0]  = o0[v] * inv;
        Or[16] = o1[v] * inv;
        Or[32] = o2[v] * inv;
        Or[48] = o3[v] * inv;
    }
}

// ---------------------------------------------------------------------------
extern "C" void kernel_launch(void* const* d_in, const int* in_sizes, int n_in,
                              void* d_out, int out_size, void* d_ws, size_t ws_size,
                              hipStream_t stream) {
    const int*   adj = (const int*)  d_in[0];
    const float* h   = (const float*)d_in[1];
    const float* Wq  = (const float*)d_in[2];
    const float* bq  = (const float*)d_in[3];
    const float* Wk  = (const float*)d_in[4];
    const float* bk  = (const float*)d_in[5];
    const float* Wv  = (const float*)d_in[6];
    const float* bv  = (const float*)d_in[7];
    float* out = (float*)d_out;

    float* qws = (float*)d_ws;                         // 16 MB
    float* kws = qws + (size_t)M_TOT * QKV_N;          // 16 MB
    float* vws = kws + (size_t)M_TOT * QKV_N;          // 16 MB

    dim3 g1(QKV_N / 16, M_TOT / (32 * 8), 3);
    qkv_gemm_kernel<<<g1, 256, 0, stream>>>(h, Wq, bq, Wk, bk, Wv, bv,
                                            qws, kws, vws);

    dim3 g2(NN / 16 / K2_WAVES, NUM_HEADS, BATCH);
    attention_kernel<<<g2, 32 * K2_WAVES, 0, stream>>>(qws, kws, vws, adj, out);
}